// SigKerMMDDiscriminator_28664611733742
// MI455X (gfx1250) — compile-verified
//
#include <hip/hip_runtime.h>

typedef float v2f  __attribute__((ext_vector_type(2)));
typedef float v8f  __attribute__((ext_vector_type(8)));

constexpr int BATCH = 64;
constexpr int STREAM = 32;
constexpr int CH = 5;
constexpr int CPAD = 8;                    // padded channel stride (zeros in 5..7)
constexpr int NPAIR = 3 * BATCH * BATCH;   // 12288 PDE problems
constexpr int DN = STREAM - 1;             // 31 increment rows/cols
constexpr int WAVES_PER_BLOCK = 8;

// ---------------------------------------------------------------------------
// Kernel A: per-pair RBF Gram via f32 WMMA, then 31x31 increments to scratch.
// One wave32 per (a,b) pair; 8 waves per 256-thread block. Stream tiles are
// staged (sigma-scaled, zero-padded to K=8) in LDS so WMMA fragments are
// unconditional ds_load_b64's — no divergent predication on the feed path.
// ---------------------------------------------------------------------------
__global__ __launch_bounds__(256) void gram_diff_kernel(
    const float* __restrict__ x, const float* __restrict__ y,
    const float* __restrict__ sigma, float* __restrict__ diff_out) {
  __shared__ float Ls[WAVES_PER_BLOCK][STREAM][CPAD];
  __shared__ float Rs[WAVES_PER_BLOCK][STREAM][CPAD];
  __shared__ float Gs[WAVES_PER_BLOCK][STREAM][STREAM + 1];
  __shared__ float nrmL[WAVES_PER_BLOCK][STREAM];
  __shared__ float nrmR[WAVES_PER_BLOCK][STREAM];

  const int wslot = threadIdx.x >> 5;
  const int lane  = threadIdx.x & 31;
  const int p     = blockIdx.x * WAVES_PER_BLOCK + wslot;   // pair id < 12288

  const int seg = p >> 12;        // 0: XX, 1: XY, 2: YY (4096 each)
  const int q   = p & 4095;
  const int a   = q >> 6;
  const int b   = q & 63;

  const float* Lp = (seg == 2) ? y : x;   // left stream source
  const float* Rp = (seg == 0) ? x : y;   // right stream source
  const bool scaleL = (seg < 2);          // mu-scaling applies to x only
  const bool scaleR = (seg == 0);

  // Channel scale factors (sigma_param applies to channels 1..4 of x).
  float sc[CH];
  sc[0] = 1.0f;
#pragma unroll
  for (int c = 1; c < CH; ++c) sc[c] = sigma[c - 1];

  // Stage: lane owns stream-row 'lane' of both sides; scale, pad, norm.
  {
    const float* lrow = Lp + (a * STREAM + lane) * CH;
    const float* rrow = Rp + (b * STREAM + lane) * CH;
    float sL = 0.0f, sR = 0.0f;
#pragma unroll
    for (int c = 0; c < CH; ++c) {
      const float u = lrow[c] * (scaleL ? sc[c] : 1.0f);
      const float w = rrow[c] * (scaleR ? sc[c] : 1.0f);
      Ls[wslot][lane][c] = u;
      Rs[wslot][lane][c] = w;
      sL += u * u;
      sR += w * w;
    }
#pragma unroll
    for (int c = CH; c < CPAD; ++c) {
      Ls[wslot][lane][c] = 0.0f;
      Rs[wslot][lane][c] = 0.0f;
    }
    nrmL[wslot][lane] = sL;
    nrmR[wslot][lane] = sR;
  }
  __syncthreads();

  const int half = lane >> 4;   // 0: lanes 0-15, 1: lanes 16-31
  const int l16  = lane & 15;

#pragma unroll
  for (int mt = 0; mt < 2; ++mt) {
#pragma unroll
    for (int nt = 0; nt < 2; ++nt) {
      v8f acc = {};
      // V_WMMA_F32_16X16X4_F32: A 16x4 f32 (2 VGPRs), B 4x16 f32 (2 VGPRs).
      // Lane layout: M/N = lane%16 (+tile*16), K = comp + 2*(lane/16) (+4*kstep).
#pragma unroll
      for (int ks = 0; ks < 2; ++ks) {
        const int c0 = 2 * half + 4 * ks;          // even -> 8B-aligned in LDS
        const int m = l16 + 16 * mt;
        const int n = l16 + 16 * nt;
        const v2f av = *(const v2f*)&Ls[wslot][m][c0];   // one ds_load_b64
        const v2f bv = *(const v2f*)&Rs[wslot][n][c0];   // one ds_load_b64
        acc = __builtin_amdgcn_wmma_f32_16x16x4_f32(
            false, av, false, bv, (short)0, acc, false, false);
      }
      // D layout: acc[r] = Dot[m = r + 8*half + 16*mt][n = l16 + 16*nt]
#pragma unroll
      for (int r = 0; r < 8; ++r) {
        const int m = r + 8 * half + 16 * mt;
        const int n = l16 + 16 * nt;
        const float d2 = nrmL[wslot][m] + nrmR[wslot][n] - 2.0f * acc[r];
        Gs[wslot][m][n] = __expf(-d2);   // RBF_SIGMA = 1
      }
    }
  }
  __syncthreads();

  // Second differences /4 (dyadic f^2 folded in); lane j < 31 owns a column.
  if (lane < DN) {
    float* outp = diff_out + (size_t)p * (DN * DN);
    for (int i = 0; i < DN; ++i) {
      const float d = 0.25f * (Gs[wslot][i + 1][lane + 1] + Gs[wslot][i][lane]
                             - Gs[wslot][i + 1][lane] - Gs[wslot][i][lane + 1]);
      outp[i * DN + lane] = d;
    }
  }
}

// ---------------------------------------------------------------------------
// Kernel B: Goursat PDE scan. One thread per problem; 63-float K row in
// registers (inner 62-column loop fully unrolled). Dyadic upsample handled by
// indexing row[j>>1] and reusing each increment row for two PDE rows.
// ---------------------------------------------------------------------------
__global__ __launch_bounds__(256) void pde_kernel(
    const float* __restrict__ diff, float* __restrict__ Kout) {
  const int t = blockIdx.x * blockDim.x + threadIdx.x;
  if (t >= NPAIR) return;
  const float* dp = diff + (size_t)t * (DN * DN);

  float K[63];
#pragma unroll
  for (int j = 0; j < 63; ++j) K[j] = 1.0f;

  for (int i2 = 0; i2 < DN; ++i2) {        // 31 increment rows
    float rowv[DN];
    const float* row = dp + i2 * DN;
#pragma unroll
    for (int jj = 0; jj < DN; ++jj) rowv[jj] = row[jj];

#pragma unroll
    for (int rep = 0; rep < 2; ++rep) {    // each increment row -> 2 PDE rows
      float kleft = 1.0f;
      float kdiag = K[0];
#pragma unroll
      for (int j = 0; j < 62; ++j) {
        const float aa   = rowv[j >> 1];
        const float kup  = K[j + 1];
        const float a212 = aa * aa * (1.0f / 12.0f);
        const float t1   = 1.0f + 0.5f * aa + a212;
        const float t2   = 1.0f - a212;
        const float knew = (kleft + kup) * t1 - kdiag * t2;
        K[j + 1] = knew;
        kdiag = kup;
        kleft = knew;
      }
    }
  }
  Kout[t] = K[62];
}

// ---------------------------------------------------------------------------
// Kernel C: weighted mean  (mean XX - 2 mean XY + mean YY), each mean /4096.
// ---------------------------------------------------------------------------
__global__ __launch_bounds__(256) void reduce_kernel(
    const float* __restrict__ Kvals, float* __restrict__ out) {
  __shared__ float sm[256];
  float acc = 0.0f;
  for (int i = threadIdx.x; i < NPAIR; i += 256) {
    const float w = (i < 4096) ? 1.0f : ((i < 8192) ? -2.0f : 1.0f);
    acc += w * Kvals[i];
  }
  sm[threadIdx.x] = acc;
  __syncthreads();
  for (int s = 128; s > 0; s >>= 1) {
    if (threadIdx.x < s) sm[threadIdx.x] += sm[threadIdx.x + s];
    __syncthreads();
  }
  if (threadIdx.x == 0) out[0] = sm[0] * (1.0f / 4096.0f);
}

extern "C" void kernel_launch(void* const* d_in, const int* in_sizes, int n_in,
                              void* d_out, int out_size, void* d_ws, size_t ws_size,
                              hipStream_t stream) {
  const float* x     = (const float*)d_in[0];
  const float* y     = (const float*)d_in[1];
  const float* sigma = (const float*)d_in[2];
  float* out = (float*)d_out;

  float* ws_diff = (float*)d_ws;                              // 12288*961 f32 (~47 MB, L2-resident)
  float* ws_K    = ws_diff + (size_t)NPAIR * (DN * DN);       // 12288 f32

  gram_diff_kernel<<<NPAIR / WAVES_PER_BLOCK, 256, 0, stream>>>(x, y, sigma, ws_diff);
  pde_kernel<<<(NPAIR + 255) / 256, 256, 0, stream>>>(ws_diff, ws_K);
  reduce_kernel<<<1, 256, 0, stream>>>(ws_K, out);
}